// AtomicBaseUpdateBlock_49091476194031
// MI455X (gfx1250) — compile-verified
//
#include <hip/hip_runtime.h>

// ---------------------------------------------------------------------------
// MI455X (gfx1250) fused MACE-style interaction block.
//  - edge kernel: fused MLP1+MLP2 (f32 WMMA 16x16x4) -> message -> atomic scatter
//  - node kernel: one-hot-selected skip contraction (VALU, LDS-staged feats)
// ---------------------------------------------------------------------------

typedef __attribute__((ext_vector_type(2))) float v2f;
typedef __attribute__((ext_vector_type(8))) float v8f;

#define N_NODES 20000
#define N_EDGES 320000
#define MULC    128
#define NEC     10
#define HS      66          // padded LDS stride for 16x64 hidden tiles

#define INV_SQRT2 0.7071067811865475f
#define INV_SQRT3 0.5773502691896258f
#define S8   0.35355339059327373f   // 1/sqrt(8)
#define S20  0.22360679774997896f   // 1/sqrt(20)
#define S64  0.125f                 // 1/sqrt(64)
#define FAN  0.02795084971874737f   // 1/sqrt(128*10)
#define INV_AVG 0.1f                // 1/AVG_NEIGH

__device__ __forceinline__ float silu_f(float v) {
    return v / (1.0f + __expf(-v));
}

// D = A(16x4) * B(4x16) + C, fp32 WMMA (wave32)
__device__ __forceinline__ v8f wmma4(v2f a, v2f b, v8f c) {
    return __builtin_amdgcn_wmma_f32_16x16x4_f32(false, a, false, b, (short)0, c,
                                                 false, false);
}

// 16x64 @ 64x64 (+silu) with hidden tile resident in LDS (A layout on read,
// C layout on write; per-wave private region, DS ops are in-order per wave).
__device__ __forceinline__ void dense64_silu(float* hp, const float* wt,
                                             int lo, int kk) {
    v2f a[16];
#pragma unroll
    for (int ks = 0; ks < 16; ++ks)
        a[ks] = *(const v2f*)(&hp[lo * HS + 4 * ks + 2 * kk]);
    v8f c4[4];
#pragma unroll
    for (int nb = 0; nb < 4; ++nb) {
        v8f c = {};
#pragma unroll
        for (int ks = 0; ks < 16; ++ks) {
            v2f b = *(const v2f*)(&wt[(nb * 16 + lo) * 64 + 4 * ks + 2 * kk]);
            c = wmma4(a[ks], b, c);
        }
        c4[nb] = c;
    }
#pragma unroll
    for (int nb = 0; nb < 4; ++nb)
#pragma unroll
        for (int r = 0; r < 8; ++r)
            hp[(r + 8 * kk) * HS + nb * 16 + lo] = silu_f(c4[nb][r]);
}

__launch_bounds__(256)
__global__ void edge_msg_kernel(
    const float* __restrict__ node_feats,
    const float* __restrict__ edge_attrs,
    const float* __restrict__ edge_feats,
    const float* __restrict__ xin,
    const int*   __restrict__ edge_index,
    const float* __restrict__ w1_0, const float* __restrict__ w2_0,
    const float* __restrict__ w1_1, const float* __restrict__ w2_1,
    const float* __restrict__ w1_2, const float* __restrict__ w2_2,
    const float* __restrict__ w1_3, const float* __restrict__ w2_3,
    float* __restrict__ bases)
{
    __shared__ float sW10[64 * 8];      // WT[n][k], scale folded
    __shared__ float sW20[64 * 20];
    __shared__ float sWA[64 * 64];      // layer-l weights, restaged per layer
    __shared__ float sWB[64 * 64];
    __shared__ float sW3m[4 * 16 * 64]; // final-layer chunk (one MLP at a time)
    __shared__ float sH[8][2][16 * HS]; // per-wave hidden tiles (mlp1, mlp2)

    const int tid  = threadIdx.x;
    const int wave = tid >> 5;
    const int lane = tid & 31;
    const int lo   = lane & 15;   // M (A-frag) / N (B-frag) lane index
    const int kk   = lane >> 4;   // K half-select

    // ---------------- stage layer-0 weights (transposed, fan folded) -------
    for (int i = tid; i < 64 * 8;  i += 256) sW10[i] = w1_0[(i & 7)  * 64 + (i >> 3)] * S8;
    for (int i = tid; i < 64 * 20; i += 256) sW20[i] = w2_0[(i % 20) * 64 + (i / 20)] * S20;
    __syncthreads();

    const long tileBase = (long)blockIdx.x * 128 + wave * 16;
    const long eRowA    = tileBase + lo;

    // ---------------- per-edge message metadata (per C-row r) --------------
    const int* sender   = edge_index;
    const int* receiver = edge_index + N_EDGES;
    float y0v[8], y1x[8], y1y[8], y1z[8];
    int   rcv[8], snd[8];
#pragma unroll
    for (int r = 0; r < 8; ++r) {
        long e = tileBase + r + 8 * kk;
        const float4 ea = *(const float4*)(edge_attrs + e * 4);
        y0v[r] = ea.x; y1x[r] = ea.y; y1y[r] = ea.z; y1z[r] = ea.w;
        rcv[r] = receiver[e];
        snd[r] = sender[e];
    }

    // ---------------- layer 0: MLP1 (K=8) and MLP2 (K=20) ------------------
    {
        const float* arow = edge_feats + eRowA * 8;
        v2f a0 = *(const v2f*)(arow + 0 + 2 * kk);
        v2f a1 = *(const v2f*)(arow + 4 + 2 * kk);
        float* hp = &sH[wave][0][0];
#pragma unroll
        for (int nb = 0; nb < 4; ++nb) {
            v8f c = {};
            v2f b0 = *(const v2f*)(&sW10[(nb * 16 + lo) * 8 + 0 + 2 * kk]);
            v2f b1 = *(const v2f*)(&sW10[(nb * 16 + lo) * 8 + 4 + 2 * kk]);
            c = wmma4(a0, b0, c);
            c = wmma4(a1, b1, c);
#pragma unroll
            for (int r = 0; r < 8; ++r)
                hp[(r + 8 * kk) * HS + nb * 16 + lo] = silu_f(c[r]);
        }
    }
    {
        const float* arow = xin + eRowA * 20;
        v2f a[5];
#pragma unroll
        for (int ks = 0; ks < 5; ++ks)
            a[ks] = *(const v2f*)(arow + 4 * ks + 2 * kk);
        float* hp = &sH[wave][1][0];
#pragma unroll
        for (int nb = 0; nb < 4; ++nb) {
            v8f c = {};
#pragma unroll
            for (int ks = 0; ks < 5; ++ks) {
                v2f b = *(const v2f*)(&sW20[(nb * 16 + lo) * 20 + 4 * ks + 2 * kk]);
                c = wmma4(a[ks], b, c);
            }
#pragma unroll
            for (int r = 0; r < 8; ++r)
                hp[(r + 8 * kk) * HS + nb * 16 + lo] = silu_f(c[r]);
        }
    }
    __syncthreads();

    // ---------------- layers 1 and 2 (64->64, silu) -------------------------
    {
        for (int i = tid; i < 4096; i += 256) sWA[i] = w1_1[(i & 63) * 64 + (i >> 6)] * S64;
        for (int i = tid; i < 4096; i += 256) sWB[i] = w2_1[(i & 63) * 64 + (i >> 6)] * S64;
        __syncthreads();
        dense64_silu(&sH[wave][0][0], sWA, lo, kk);
        dense64_silu(&sH[wave][1][0], sWB, lo, kk);
        __syncthreads();
        for (int i = tid; i < 4096; i += 256) sWA[i] = w1_2[(i & 63) * 64 + (i >> 6)] * S64;
        for (int i = tid; i < 4096; i += 256) sWB[i] = w2_2[(i & 63) * 64 + (i >> 6)] * S64;
        __syncthreads();
        dense64_silu(&sH[wave][0][0], sWA, lo, kk);
        dense64_silu(&sH[wave][1][0], sWB, lo, kk);
    }

    // ---------------- layer 3 (64->512) fused with message scatter ---------
#pragma unroll 1
    for (int ch = 0; ch < 8; ++ch) {
        const int c0 = ch * 16;
        v8f ct[4];
        {
            v8f z = {};
            ct[0] = z; ct[1] = z; ct[2] = z; ct[3] = z;
        }
#pragma unroll 1
        for (int mlp = 0; mlp < 2; ++mlp) {
            const float* src = mlp ? w2_3 : w1_3;
            __syncthreads();   // prior users of sW3m are done
            for (int i = tid; i < 4096; i += 256) {
                int nn = i & 15, k = (i >> 4) & 63, t = i >> 10;
                sW3m[(t * 16 + nn) * 64 + k] = src[k * 512 + t * 128 + c0 + nn] * S64;
            }
            __syncthreads();
            if (ch + 1 < 8)   // warm L2/L0 for next chunk's weight columns
                __builtin_prefetch(src + (tid & 63) * 512 + (ch + 1) * 16, 0, 1);
            const float* hp = &sH[wave][mlp][0];
            v2f a[16];
#pragma unroll
            for (int ks = 0; ks < 16; ++ks)
                a[ks] = *(const v2f*)(&hp[lo * HS + 4 * ks + 2 * kk]);
#pragma unroll
            for (int ks = 0; ks < 16; ++ks) {
#pragma unroll
                for (int t = 0; t < 4; ++t) {
                    v2f b = *(const v2f*)(&sW3m[(t * 16 + lo) * 64 + 4 * ks + 2 * kk]);
                    ct[t] = wmma4(a[ks], b, ct[t]);
                }
            }
        }
        // each lane owns channel c = c0+lo of edges r+8*kk (C-tile layout)
        const int c = c0 + lo;
#pragma unroll
        for (int r = 0; r < 8; ++r) {
            const float* nf  = node_feats + (long)snd[r] * 512;
            float xs0 = nf[c];
            const float* x1p = nf + 128 + 3 * c;
            float x1a = x1p[0], x1b = x1p[1], x1c = x1p[2];
            float dot = x1a * y1x[r] + x1b * y1y[r] + x1c * y1z[r];
            float wav = ct[0][r], wbv = ct[1][r], wcv = ct[2][r], wdv = ct[3][r];
            float m0  = INV_SQRT2 * (wav * xs0 * y0v[r] + wdv * dot * INV_SQRT3);
            float m1a = INV_SQRT2 * (wbv * xs0 * y1x[r] + wcv * x1a * y0v[r]);
            float m1b = INV_SQRT2 * (wbv * xs0 * y1y[r] + wcv * x1b * y0v[r]);
            float m1c = INV_SQRT2 * (wbv * xs0 * y1z[r] + wcv * x1c * y0v[r]);
            float* bout = bases + (long)rcv[r] * 512;
            atomicAdd(bout + c,               m0  * INV_AVG);
            atomicAdd(bout + 128 + 3 * c + 0, m1a * INV_AVG);
            atomicAdd(bout + 128 + 3 * c + 1, m1b * INV_AVG);
            atomicAdd(bout + 128 + 3 * c + 2, m1c * INV_AVG);
        }
    }
}

// ---------------------------------------------------------------------------
// Per-node one-hot-selected skip contraction:
//   sc0[n,k]   = fan * sum_u x0[n,u]   * skip_w0[u, v_n, k]
//   sc1[n,k,m] = fan * sum_u x1[n,u,m] * skip_w1[u, v_n, k]
// ---------------------------------------------------------------------------
__launch_bounds__(256)
__global__ void sc_kernel(const float* __restrict__ node_attrs,
                          const float* __restrict__ node_feats,
                          const float* __restrict__ skip_w0,
                          const float* __restrict__ skip_w1,
                          float* __restrict__ sc)
{
    __shared__ float xf[512];
    __shared__ int vsh;
    const int n   = blockIdx.x;
    const int tid = threadIdx.x;
    for (int i = tid; i < 512; i += 256) xf[i] = node_feats[(long)n * 512 + i];
    if (tid < NEC && node_attrs[(long)n * NEC + tid] > 0.5f) vsh = tid;
    __syncthreads();
    const int v = vsh;
    for (int o = tid; o < 512; o += 256) {
        float s = 0.0f;
        if (o < MULC) {
            const float* w = skip_w0 + (long)v * MULC + o;   // stride NEC*MULC over u
#pragma unroll 8
            for (int u = 0; u < MULC; ++u) s += xf[u] * w[(long)u * (NEC * MULC)];
        } else {
            int j = o - MULC, k = j / 3, m = j - 3 * k;
            const float* w = skip_w1 + (long)v * MULC + k;
#pragma unroll 8
            for (int u = 0; u < MULC; ++u) s += xf[MULC + 3 * u + m] * w[(long)u * (NEC * MULC)];
        }
        sc[(long)n * 512 + o] = s * FAN;
    }
}

extern "C" void kernel_launch(void* const* d_in, const int* in_sizes, int n_in,
                              void* d_out, int out_size, void* d_ws, size_t ws_size,
                              hipStream_t stream) {
    const float* node_attrs = (const float*)d_in[0];
    const float* node_feats = (const float*)d_in[1];
    const float* edge_attrs = (const float*)d_in[2];
    const float* edge_feats = (const float*)d_in[3];
    const float* xin        = (const float*)d_in[4];
    const int*   edge_index = (const int*)d_in[5];
    // d_in[6] = node_num scalar (fixed at N_NODES)
    const float* w1_0 = (const float*)d_in[7];
    const float* w2_0 = (const float*)d_in[8];
    const float* w1_1 = (const float*)d_in[9];
    const float* w2_1 = (const float*)d_in[10];
    const float* w1_2 = (const float*)d_in[11];
    const float* w2_2 = (const float*)d_in[12];
    const float* w1_3 = (const float*)d_in[13];
    const float* w2_3 = (const float*)d_in[14];
    const float* skip_w0 = (const float*)d_in[15];
    const float* skip_w1 = (const float*)d_in[16];

    float* bases = (float*)d_out;
    float* sc    = (float*)d_out + (size_t)N_NODES * 512;

    hipMemsetAsync(bases, 0, (size_t)N_NODES * 512 * sizeof(float), stream);

    sc_kernel<<<N_NODES, 256, 0, stream>>>(node_attrs, node_feats,
                                           skip_w0, skip_w1, sc);

    edge_msg_kernel<<<N_EDGES / 128, 256, 0, stream>>>(
        node_feats, edge_attrs, edge_feats, xin, edge_index,
        w1_0, w2_0, w1_1, w2_1, w1_2, w2_2, w1_3, w2_3, bases);
}